// EnhancedMattingLoss_14955076125159
// MI455X (gfx1250) — compile-verified
//
#include <hip/hip_runtime.h>
#include <math.h>

// ---------------- problem constants (match reference setup_inputs) ----------
#define BATCH   2
#define HH      384
#define WW      384
#define KWIN    11            // consistency kernel (11x11 window, top-11)
#define PAD     5             // KWIN/2
#define KK2     (KWIN*KWIN)   // 121 neighbors
#define TOPK    11

#define TILE_W  32
#define TILE_H  8
#define LTW     (TILE_W + 2*PAD)   // 42
#define LTH     (TILE_H + 2*PAD)   // 18
#define LELEMS  (LTW*LTH)          // 756
#define NTHREADS 256

#define KNOWN_WEIGHT 1.0f
#define DDC_WEIGHT   0.1f

// ---------------- CDNA5 async global->LDS staging ---------------------------
#if defined(__gfx1250__) && __has_builtin(__builtin_amdgcn_global_load_async_to_lds_b32)
#define HAVE_ASYNC_LDS 1
#else
#define HAVE_ASYNC_LDS 0
#endif

typedef __attribute__((address_space(1))) int glb_i32;
typedef __attribute__((address_space(3))) int lds_i32;

__device__ __forceinline__ void async_g2l_b32(const float* g, float* l) {
#if HAVE_ASYNC_LDS
  // emits global_load_async_to_lds_b32 (tracked by ASYNCcnt)
  __builtin_amdgcn_global_load_async_to_lds_b32(
      (glb_i32*)(g),
      (lds_i32*)(l),
      /*imm offset*/0, /*cpol*/0);
#else
  *l = *g;
#endif
}

__device__ __forceinline__ void wait_async_all() {
#if HAVE_ASYNC_LDS
#if __has_builtin(__builtin_amdgcn_s_wait_asynccnt)
  __builtin_amdgcn_s_wait_asynccnt(0);
#else
  asm volatile("s_wait_asynccnt 0" ::: "memory");
#endif
#endif
}

// ---------------- main tile kernel ------------------------------------------
__global__ __launch_bounds__(NTHREADS)
void matting_loss_tile_kernel(const float* __restrict__ alpha,
                              const float* __restrict__ trimap,
                              const float* __restrict__ image,
                              float* __restrict__ partials) {
  __shared__ float sR[LELEMS];
  __shared__ float sG[LELEMS];
  __shared__ float sB[LELEMS];
  __shared__ float sA[LELEMS];
  __shared__ float red[NTHREADS * 4];

  const int bx = blockIdx.x, by = blockIdx.y, b = blockIdx.z;
  const int tid = threadIdx.x;
  const int x0 = bx * TILE_W - PAD;
  const int y0 = by * TILE_H - PAD;

  const size_t planeHW = (size_t)HH * WW;
  const float* imgR = image + ((size_t)b * 3 + 0) * planeHW;
  const float* imgG = image + ((size_t)b * 3 + 1) * planeHW;
  const float* imgB = image + ((size_t)b * 3 + 2) * planeHW;
  const float* alp  = alpha  + (size_t)b * planeHW;
  const float* tri  = trimap + (size_t)b * planeHW;

  // ---- stage halo tile: async global->LDS for in-bounds, zeros for padding.
  // Zero-pad is applied to the *de-normalized* image in the reference, so
  // pad cells must stay exactly 0 after the in-place denormalize below.
  for (int e = tid; e < LELEMS; e += NTHREADS) {
    const int ty = e / LTW, tx = e - ty * LTW;
    const int gy = y0 + ty, gx = x0 + tx;
    const bool inb = (gy >= 0) & (gy < HH) & (gx >= 0) & (gx < WW);
    if (inb) {
      const size_t g = (size_t)gy * WW + gx;
      async_g2l_b32(imgR + g, &sR[e]);
      async_g2l_b32(imgG + g, &sG[e]);
      async_g2l_b32(imgB + g, &sB[e]);
      async_g2l_b32(alp  + g, &sA[e]);
    } else {
      sR[e] = 0.0f; sG[e] = 0.0f; sB[e] = 0.0f; sA[e] = 0.0f;
    }
  }
  wait_async_all();
  __syncthreads();

  // ---- de-normalize in place (in-bounds cells only; pads stay 0)
  for (int e = tid; e < LELEMS; e += NTHREADS) {
    const int ty = e / LTW, tx = e - ty * LTW;
    const int gy = y0 + ty, gx = x0 + tx;
    if ((gy >= 0) & (gy < HH) & (gx >= 0) & (gx < WW)) {
      sR[e] = fmaf(sR[e], 0.229f, 0.485f);
      sG[e] = fmaf(sG[e], 0.224f, 0.456f);
      sB[e] = fmaf(sB[e], 0.225f, 0.406f);
    }
  }
  __syncthreads();

  // ---- per-pixel work
  const int tx = tid % TILE_W;
  const int ty = tid / TILE_W;
  const int gx = bx * TILE_W + tx;
  const int gy = by * TILE_H + ty;
  const size_t g = (size_t)gy * WW + gx;

  const int c0 = (ty + PAD) * LTW + (tx + PAD);
  const float a = sA[c0];
  const float t = tri[g];

  // known-region L1 term
  const bool known = (t != 0.5f);
  float kl = known ? fabsf(a - t) : 0.0f;
  float kc = known ? 1.0f : 0.0f;

  // DDC consistency term (only where trimap == 0.5)
  float dsum = 0.0f;
  float sc = 0.0f;
  if (t == 0.5f) {
    sc = 1.0f;
    const float cR = sR[c0], cG = sG[c0], cB = sB[c0];

    // stable 11-smallest-of-121 selection, kept entirely in registers.
    float bd[TOPK];
    int   bi[TOPK];
#pragma unroll
    for (int j = 0; j < TOPK; ++j) { bd[j] = 3.0e38f; bi[j] = 0x7fffffff; }

    for (int dy = 0; dy < KWIN; ++dy) {
      const int row = (ty + dy) * LTW + tx;
      for (int dx = 0; dx < KWIN; ++dx) {
        const int o = row + dx;
        const float dR = cR - sR[o];
        const float dG = cG - sG[o];
        const float dB = cB - sB[o];
        float d = sqrtf(fmaf(dR, dR, fmaf(dG, dG, dB * dB)));
        int id = dy * KWIN + dx;
        // strict-< insert while scanning in increasing index order ==
        // jax.lax.top_k stable tie-breaking (lowest index wins ties).
#pragma unroll
        for (int j = 0; j < TOPK; ++j) {
          const bool sw = d < bd[j];
          const float td = bd[j]; const int ti = bi[j];
          bd[j] = sw ? d : td;  bi[j] = sw ? id : ti;
          d     = sw ? td : d;  id    = sw ? ti : id;
        }
      }
    }
    // gather alpha diffs at the selected neighbor indices
#pragma unroll
    for (int j = 0; j < TOPK; ++j) {
      const int id = bi[j];
      const int dy = id / KWIN;
      const int dx = id - dy * KWIN;
      const float na = sA[(ty + dy) * LTW + (tx + dx)];
      dsum += fabsf(bd[j] - (a - na));
    }
  }

  // ---- deterministic block reduction of 4 quantities
  red[tid * 4 + 0] = kl;
  red[tid * 4 + 1] = kc;
  red[tid * 4 + 2] = dsum;
  red[tid * 4 + 3] = sc;
  __syncthreads();
  for (int s = NTHREADS / 2; s > 0; s >>= 1) {
    if (tid < s) {
      red[tid * 4 + 0] += red[(tid + s) * 4 + 0];
      red[tid * 4 + 1] += red[(tid + s) * 4 + 1];
      red[tid * 4 + 2] += red[(tid + s) * 4 + 2];
      red[tid * 4 + 3] += red[(tid + s) * 4 + 3];
    }
    __syncthreads();
  }
  if (tid == 0) {
    const int blk = ((int)blockIdx.z * gridDim.y + blockIdx.y) * gridDim.x + blockIdx.x;
    partials[blk * 4 + 0] = red[0];
    partials[blk * 4 + 1] = red[1];
    partials[blk * 4 + 2] = red[2];
    partials[blk * 4 + 3] = red[3];
  }
}

// ---------------- finalize: fixed-order reduction + loss composition --------
__global__ __launch_bounds__(NTHREADS)
void matting_loss_finalize_kernel(const float* __restrict__ partials,
                                  int nblk, float* __restrict__ out) {
  __shared__ float red[NTHREADS * 4];
  const int tid = threadIdx.x;
  float s0 = 0.0f, s1 = 0.0f, s2 = 0.0f, s3 = 0.0f;
  for (int i = tid; i < nblk; i += NTHREADS) {
    s0 += partials[i * 4 + 0];
    s1 += partials[i * 4 + 1];
    s2 += partials[i * 4 + 2];
    s3 += partials[i * 4 + 3];
  }
  red[tid * 4 + 0] = s0; red[tid * 4 + 1] = s1;
  red[tid * 4 + 2] = s2; red[tid * 4 + 3] = s3;
  __syncthreads();
  for (int s = NTHREADS / 2; s > 0; s >>= 1) {
    if (tid < s) {
      red[tid * 4 + 0] += red[(tid + s) * 4 + 0];
      red[tid * 4 + 1] += red[(tid + s) * 4 + 1];
      red[tid * 4 + 2] += red[(tid + s) * 4 + 2];
      red[tid * 4 + 3] += red[(tid + s) * 4 + 3];
    }
    __syncthreads();
  }
  if (tid == 0) {
    const float kl_sum = red[0];
    const float kcnt   = red[1];
    const float dsum   = red[2];
    const float scnt   = red[3];
    const float inv_known = 1.0f / ((float)BATCH * HH * WW);                 // mean over [b,1,h,w]
    const float inv_ddc   = 1.0f / ((float)BATCH * TOPK * (float)HH * WW);   // mean over [b,k,h,w]
    const float known_l1 = (kcnt > 0.0f) ? kl_sum * inv_known * KNOWN_WEIGHT : 0.0f;
    const float ddc      = (scnt > 0.0f) ? dsum * inv_ddc * DDC_WEIGHT : 0.0f;
    out[0] = known_l1 + ddc;  // total
    out[1] = known_l1;
    out[2] = ddc;
  }
}

// ---------------- host entry -------------------------------------------------
extern "C" void kernel_launch(void* const* d_in, const int* in_sizes, int n_in,
                              void* d_out, int out_size, void* d_ws, size_t ws_size,
                              hipStream_t stream) {
  const float* pred_alpha  = (const float*)d_in[0];  // [2,1,384,384]
  const float* gt_trimap   = (const float*)d_in[1];  // [2,1,384,384]
  const float* input_image = (const float*)d_in[2];  // [2,3,384,384]
  float* out = (float*)d_out;                         // 3 floats
  float* partials = (float*)d_ws;                     // nblk*4 floats (every slot rewritten each call)

  dim3 grid(WW / TILE_W, HH / TILE_H, BATCH);         // 12 x 48 x 2 = 1152 blocks
  matting_loss_tile_kernel<<<grid, NTHREADS, 0, stream>>>(
      pred_alpha, gt_trimap, input_image, partials);

  const int nblk = grid.x * grid.y * grid.z;
  matting_loss_finalize_kernel<<<1, NTHREADS, 0, stream>>>(partials, nblk, out);
}